// TextLevelGNN_2362232013143
// MI455X (gfx1250) — compile-verified
//
#include <hip/hip_runtime.h>
#include <math.h>

// Problem constants (from reference)
constexpr int NB = 128;   // batch
constexpr int NL = 512;   // sequence length
constexpr int NK = 8;     // neighbors
constexpr int ND = 300;   // embed dim
constexpr int NC = 14;    // classes
constexpr int LCHUNKS = 8;
constexpr int LPER = NL / LCHUNKS;   // 64

typedef __attribute__((ext_vector_type(2))) float v2f;
typedef __attribute__((ext_vector_type(8))) float v8f;

// -----------------------------------------------------------------------------
// Kernel 1: gather + neighbor max-pool + gated partial sum over an L-chunk.
//   partial[(b*LCHUNKS + chunk)*ND + d] = sum_{l in chunk} (1-Nn)*Mn + Nn*Rn
// One block per (b, chunk); thread d handles one embedding dimension.
// R table (~5.9 MB) is L2-resident; consecutive d lanes give coalesced loads
// within each gathered row.
// -----------------------------------------------------------------------------
__global__ __launch_bounds__(320)
void gnn_gather_kernel(const int* __restrict__ master,
                       const int* __restrict__ slave,
                       const int* __restrict__ edges,
                       const float* __restrict__ R,
                       const float* __restrict__ E,
                       const float* __restrict__ N,
                       float* __restrict__ partial) {
    const int b     = blockIdx.x;
    const int chunk = blockIdx.y;
    const int d     = threadIdx.x;
    if (d >= ND) return;   // no barriers below; early-exit is safe

    float acc = 0.0f;
    const int l0 = chunk * LPER;
    for (int li = 0; li < LPER; ++li) {
        const int l = l0 + li;
        const int m = master[b * NL + l];
        const float nn = N[m];

        const int* sp = slave + ((size_t)(b * NL + l)) * NK;
        const int* ep = edges + ((size_t)(b * NL + l)) * NK;

        // k = 0 initializes the running max (K is always 8, matches jnp.max)
        float ev = E[ep[0]];
        float mn = R[(size_t)sp[0] * ND + d] * ev;
#pragma unroll
        for (int k = 1; k < NK; ++k) {
            float e = E[ep[k]];
            float v = R[(size_t)sp[k] * ND + d] * e;
            mn = fmaxf(mn, v);
        }
        const float rn = R[(size_t)m * ND + d];
        acc += (1.0f - nn) * mn + nn * rn;
    }
    partial[(size_t)(b * LCHUNKS + chunk) * ND + d] = acc;
}

// -----------------------------------------------------------------------------
// Kernel 2: deterministic reduction of L-chunk partials -> x[b][d]
// -----------------------------------------------------------------------------
__global__ __launch_bounds__(320)
void gnn_reduce_kernel(const float* __restrict__ partial,
                       float* __restrict__ x) {
    const int b = blockIdx.x;
    const int d = threadIdx.x;
    if (d >= ND) return;
    float s = 0.0f;
#pragma unroll
    for (int c = 0; c < LCHUNKS; ++c)
        s += partial[(size_t)(b * LCHUNKS + c) * ND + d];
    x[(size_t)b * ND + d] = s;
}

// -----------------------------------------------------------------------------
// Kernel 3: head GEMM via V_WMMA_F32_16X16X4_F32 + bias + ReLU + softmax.
// One block, 256 threads = 8 wave32 waves; wave w computes rows [16w, 16w+16).
// W is staged zero-padded to 16 rows in LDS so the WMMA loop has no divergence
// (EXEC must be all-1s for WMMA).
//
// Lane layouts (CDNA5 ISA 7.12.2, 32-bit operands, 16x16x4):
//   A (16x4):  lane = 16*half + m ; VGPR0 = A[m][2*half], VGPR1 = A[m][2*half+1]
//   B (4x16):  lane = 16*half + n ; VGPR0 = B[2*half][n], VGPR1 = B[2*half+1][n]
//   C/D (16x16): VGPR r ; lanes 0-15: M=r, N=lane ; lanes 16-31: M=r+8, N=lane-16
// -----------------------------------------------------------------------------
__global__ __launch_bounds__(256)
void gnn_head_kernel(const float* __restrict__ x,
                     const float* __restrict__ W,     // [NC][ND] row-major
                     const float* __restrict__ bias,  // [NC]
                     float* __restrict__ out) {       // [NB][NC]
    __shared__ float Wl[16 * ND];      // zero-padded W: Wl[n*ND + k], n in [0,16)
    __shared__ float hbuf[NB][16];     // raw GEMM results

    const int tid  = threadIdx.x;
    const int lane = tid & 31;
    const int wave = tid >> 5;         // M tile: rows [wave*16, wave*16+16)
    const int half = lane >> 4;        // 0 or 1
    const int ln   = lane & 15;        // m for A loads, n for B loads / D cols

    // Stage zero-padded W into LDS
    for (int idx = tid; idx < 16 * ND; idx += 256) {
        const int n = idx / ND;
        const int k = idx % ND;
        Wl[idx] = (n < NC) ? W[n * ND + k] : 0.0f;
    }
    __syncthreads();

    const int row = wave * 16 + ln;    // A-matrix row for this lane
    v8f c = {};                        // f32 accumulator, 8 VGPRs

    for (int k0 = 0; k0 < ND; k0 += 4) {
        const int ka = k0 + (half << 1);
        v2f a, bm;
        a.x  = x[(size_t)row * ND + ka];
        a.y  = x[(size_t)row * ND + ka + 1];
        bm.x = Wl[ln * ND + ka];
        bm.y = Wl[ln * ND + ka + 1];
        // (neg_a, A, neg_b, B, c_mod, C, reuse_a, reuse_b)
        c = __builtin_amdgcn_wmma_f32_16x16x4_f32(
                false, a, false, bm, (short)0, c, false, false);
    }

    // Scatter D tile to LDS per C/D layout
#pragma unroll
    for (int r = 0; r < 8; ++r) {
        const int mrow = wave * 16 + r + (half << 3);
        hbuf[mrow][ln] = c[r];
    }
    __syncthreads();

    // Per-row bias + ReLU + softmax (one thread per batch row)
    if (tid < NB) {
        float h[NC];
        float mx = -INFINITY;
#pragma unroll
        for (int j = 0; j < NC; ++j) {
            float v = hbuf[tid][j] + bias[j];
            v = fmaxf(v, 0.0f);
            h[j] = v;
            mx = fmaxf(mx, v);
        }
        float s = 0.0f;
#pragma unroll
        for (int j = 0; j < NC; ++j) {
            h[j] = __expf(h[j] - mx);
            s += h[j];
        }
        const float inv = 1.0f / s;
#pragma unroll
        for (int j = 0; j < NC; ++j)
            out[tid * NC + j] = h[j] * inv;
    }
}

// -----------------------------------------------------------------------------
// Launch
// -----------------------------------------------------------------------------
extern "C" void kernel_launch(void* const* d_in, const int* in_sizes, int n_in,
                              void* d_out, int out_size, void* d_ws, size_t ws_size,
                              hipStream_t stream) {
    const int*   master = (const int*)  d_in[0];   // [128,512]
    const int*   slave  = (const int*)  d_in[1];   // [128,512,8]
    const int*   edges  = (const int*)  d_in[2];   // [128,512,8]
    const float* R      = (const float*)d_in[3];   // [4905,300]
    const float* E      = (const float*)d_in[4];   // [4905*4905... +1, 1]
    const float* N      = (const float*)d_in[5];   // [4905,1]
    const float* W      = (const float*)d_in[6];   // [14,300]
    const float* bias   = (const float*)d_in[7];   // [14]
    float*       out    = (float*)d_out;           // [128,14]

    // Workspace layout
    float* partial = (float*)d_ws;                              // NB*LCHUNKS*ND
    float* x       = partial + (size_t)NB * LCHUNKS * ND;       // NB*ND

    gnn_gather_kernel<<<dim3(NB, LCHUNKS), 320, 0, stream>>>(
        master, slave, edges, R, E, N, partial);

    gnn_reduce_kernel<<<NB, 320, 0, stream>>>(partial, x);

    gnn_head_kernel<<<1, 256, 0, stream>>>(x, W, bias, out);
}